// LSTMEncoder_43138651521423
// MI455X (gfx1250) — compile-verified
//
#include <hip/hip_runtime.h>

// LSTM encoder for MI455X (gfx1250): 2x LSTM (B=64,T=256,H=1024) + LayerNorm.
// fp32 end-to-end using V_WMMA_F32_16X16X4_F32 on the matrix pipe.
// Grid per step: 64 column-blocks x 2 batch-halves = 128 WGs (WGP coverage),
// 8 waves/WG = (16-row tile) x (K-quarter). Workspace: ~64.3 MB.

typedef float v2f __attribute__((ext_vector_type(2)));
typedef float v8f __attribute__((ext_vector_type(8)));

#define B_   64
#define T_   256
#define H_   1024
#define G4_  4096
#define ROWSTRIDE (T_ * H_)   // stride between batch rows in a [B,T,H] sequence
#define XPAD 18               // LDS row pitch: 8*18 % 64 == 16 -> conflict-free halves

__device__ __forceinline__ float sigm_f(float x) {
    return 1.0f / (1.0f + __expf(-x));
}
__device__ __forceinline__ float tanh_f(float x) {
    float e = __expf(-2.0f * x);
    return (1.0f - e) / (1.0f + e);
}

// One LSTM timestep: gates = [x_t | h_{t-1}] * [Wi;Wh] + b, then cell update.
// blockIdx.x = 16-wide h-column block (64), blockIdx.y = 32-row batch half (2).
// Block: 256 threads = 8 waves; wave = (mi: 16-row tile in the half, ks: K-quarter).
// K-quarters 0,1 cover x_t*Wi (K=1024); 2,3 cover h_{t-1}*Wh (K=1024).
// Each wave accumulates all 4 gate tiles so the cell update is per-lane VALU.
__global__ __launch_bounds__(256) void lstm_step_kernel(
    const float* __restrict__ xseq,   // [B,T,H] input sequence for this layer
    const float* __restrict__ Wi,     // [H,4H]
    const float* __restrict__ Wh,     // [H,4H]
    const float* __restrict__ bias,   // [4H]
    float*       __restrict__ hseq,   // [B,T,H] output hidden sequence
    float*       __restrict__ cstate, // [B,H]
    int t)
{
    // Partial-sum exchange: [ks-1][mi][gate][row][col(padded)]
    __shared__ float xfer[3][2][4][16][XPAD];

    const int tid  = threadIdx.x;
    const int lane = tid & 31;
    const int wave = tid >> 5;        // 0..7
    const int mi   = wave & 1;        // 16-row tile within the 32-row batch half
    const int ks   = wave >> 1;       // 0..3 K-quarter
    const int n    = lane & 15;       // col within tile / A-row selector
    const int h2   = lane >> 4;       // lane half (K sub-pair per WMMA layout)
    const int col0 = blockIdx.x * 16;             // h-column base (0..1008)
    const int row0 = blockIdx.y * 32 + mi * 16;   // batch-row base (0..48)

    v8f acc[4] = {};                  // i, f, g, o accumulators (16x16 f32 each)

    const bool isWh   = (ks >= 2);
    const bool doGemm = !isWh || (t > 0);
    if (doGemm) {
        const float* Asrc = isWh ? (hseq + (size_t)(t - 1) * H_)
                                 : (xseq + (size_t)t * H_);
        const float* W    = isWh ? Wh : Wi;
        const int    kb   = (ks & 1) * 512;       // K-quarter base within the matrix
        // A fragment: lane (16*h2 + m) holds A[m][kk+2*h2 .. +1]; m == n here.
        const float* Arow  = Asrc + (size_t)(row0 + n) * ROWSTRIDE + kb;
        const float* Wbase = W + (size_t)kb * G4_ + col0 + n;

        for (int kk = 0; kk < 512; kk += 4) {
            const int ka = kk + 2 * h2;
            const v2f a = *(const v2f*)(Arow + ka);   // 8B-aligned (ka even)
            // B fragment: lane (16*h2 + n) holds W[kb+ka][c], W[kb+ka+1][c]
            const float* w0 = Wbase + (size_t)ka * G4_;
            const float* w1 = w0 + G4_;
#pragma unroll
            for (int g = 0; g < 4; ++g) {
                v2f bv;
                bv.x = w0[g * H_];
                bv.y = w1[g * H_];
                acc[g] = __builtin_amdgcn_wmma_f32_16x16x4_f32(
                    /*neg_a=*/false, a, /*neg_b=*/false, bv,
                    /*c_mod=*/(short)0, acc[g],
                    /*reuse_a=*/false, /*reuse_b=*/false);
            }
        }
    }

    // K-quarters 1..3 publish partials (zeros for Wh quarters at t==0).
    if (ks != 0) {
#pragma unroll
        for (int g = 0; g < 4; ++g)
#pragma unroll
            for (int r = 0; r < 8; ++r)
                xfer[ks - 1][mi][g][r + 8 * h2][n] = acc[g][r];
    }
    __syncthreads();

    if (ks == 0) {
#pragma unroll
        for (int g = 0; g < 4; ++g) {
            const float bg = bias[g * H_ + col0 + n];
#pragma unroll
            for (int r = 0; r < 8; ++r) {
                acc[g][r] += xfer[0][mi][g][r + 8 * h2][n]
                           + xfer[1][mi][g][r + 8 * h2][n]
                           + xfer[2][mi][g][r + 8 * h2][n] + bg;
            }
        }
        // Pointwise LSTM cell: same lane/slot holds matching i,f,g,o elements.
#pragma unroll
        for (int r = 0; r < 8; ++r) {
            const int brow = row0 + r + 8 * h2;   // batch index
            const int colh = col0 + n;            // hidden index
            const float iv = sigm_f(acc[0][r]);
            const float fv = sigm_f(acc[1][r]);
            const float gv = tanh_f(acc[2][r]);
            const float ov = sigm_f(acc[3][r]);
            const float cp = (t > 0) ? cstate[brow * H_ + colh] : 0.0f;
            const float cn = fv * cp + iv * gv;
            cstate[brow * H_ + colh] = cn;
            hseq[(size_t)brow * ROWSTRIDE + (size_t)t * H_ + colh] = ov * tanh_f(cn);
        }
    }
}

// In-place LayerNorm over the last dim (H=1024). One block per (b,t) row.
__global__ __launch_bounds__(256) void layernorm_kernel(
    float* __restrict__ out,
    const float* __restrict__ scale,
    const float* __restrict__ bias)
{
    __shared__ float red[256];
    float* p = out + (size_t)blockIdx.x * H_;
    const int tid = threadIdx.x;

    float4 x = *(const float4*)(p + tid * 4);

    red[tid] = x.x + x.y + x.z + x.w;
    __syncthreads();
    for (int off = 128; off > 0; off >>= 1) {
        if (tid < off) red[tid] += red[tid + off];
        __syncthreads();
    }
    const float mu = red[0] * (1.0f / H_);
    __syncthreads();

    const float dx = x.x - mu, dy = x.y - mu, dz = x.z - mu, dw = x.w - mu;
    red[tid] = dx * dx + dy * dy + dz * dz + dw * dw;
    __syncthreads();
    for (int off = 128; off > 0; off >>= 1) {
        if (tid < off) red[tid] += red[tid + off];
        __syncthreads();
    }
    const float inv = rsqrtf(red[0] * (1.0f / H_) + 1e-6f);

    const float4 sc = *(const float4*)(scale + tid * 4);
    const float4 bs = *(const float4*)(bias + tid * 4);
    float4 r;
    r.x = dx * inv * sc.x + bs.x;
    r.y = dy * inv * sc.y + bs.y;
    r.z = dz * inv * sc.z + bs.z;
    r.w = dw * inv * sc.w + bs.w;
    *(float4*)(p + tid * 4) = r;
}

extern "C" void kernel_launch(void* const* d_in, const int* in_sizes, int n_in,
                              void* d_out, int out_size, void* d_ws, size_t ws_size,
                              hipStream_t stream)
{
    const float* x        = (const float*)d_in[0];
    const float* Wi0      = (const float*)d_in[1];
    const float* Wh0      = (const float*)d_in[2];
    const float* b0       = (const float*)d_in[3];
    const float* Wi1      = (const float*)d_in[4];
    const float* Wh1      = (const float*)d_in[5];
    const float* b1       = (const float*)d_in[6];
    const float* ln_scale = (const float*)d_in[7];
    const float* ln_bias  = (const float*)d_in[8];
    float* out = (float*)d_out;

    // Workspace layout: h0 sequence [B,T,H], then c state [B,H] (reused by both layers).
    float* h0buf = (float*)d_ws;
    float* cbuf  = h0buf + (size_t)B_ * T_ * H_;

    const dim3 blk(256);
    const dim3 grid(H_ / 16, 2);   // 64 column blocks x 2 batch halves = 128 WGs

    // Layer 0: x -> h0buf
    for (int t = 0; t < T_; ++t)
        lstm_step_kernel<<<grid, blk, 0, stream>>>(x, Wi0, Wh0, b0, h0buf, cbuf, t);
    // Layer 1: h0buf -> out
    for (int t = 0; t < T_; ++t)
        lstm_step_kernel<<<grid, blk, 0, stream>>>(h0buf, Wi1, Wh1, b1, out, cbuf, t);
    // LayerNorm in place on out
    layernorm_kernel<<<dim3(B_ * T_), blk, 0, stream>>>(out, ln_scale, ln_bias);
}